// SparseLinear_43379169690256
// MI455X (gfx1250) — compile-verified
//
#include <hip/hip_runtime.h>

// CDNA5 / gfx1250: wave32, WMMA f32 16x16x4.
typedef __attribute__((ext_vector_type(2))) float v2f;
typedef __attribute__((ext_vector_type(8))) float v8f;

#define B_  256
#define K_  500
#define D_  512
#define KT_ 32   // ceil(K/16) k-tiles per batch row

// One wave (32 threads) per (b, 16-k tile).
// A operand: 16 gathered weight rows, per-lane 8B loads in the ISA A-layout
//   (lane m -> row_m K=0..1, lane m+16 -> row_m K=2..3  => adjacent 8B).
// B operand: embed[b] chunk broadcast across the 16 N columns, read from LDS
//   with the mirrored layout. Only column 0 of D is used.
__global__ __launch_bounds__(32) void sparse_linear_wmma(
    const float* __restrict__ embed,
    const long long* __restrict__ shortlist,
    const float* __restrict__ weight,
    const float* __restrict__ bias,
    float* __restrict__ out)
{
    __shared__ float s_e[D_];     // staged embed row (2 KB)
    __shared__ float s_col[16];   // D-matrix column 0

    const int lane = threadIdx.x;      // 0..31
    const int b    = blockIdx.y;
    const int kb   = blockIdx.x * 16;
    const int m    = lane & 15;
    const int half = lane >> 4;        // 0 or 1

    // Stage embed[b,:] into LDS: 512 floats = 4x float4 per lane, coalesced.
    {
        const float4* esrc = (const float4*)(embed + (size_t)b * D_);
        float4* edst = (float4*)s_e;
#pragma unroll
        for (int i = 0; i < 4; ++i)
            edst[lane + 32 * i] = esrc[lane + 32 * i];
    }
    __syncthreads();

    int k = kb + m;
    if (k >= K_) k = K_ - 1;           // clamp tail tile (dup work, store guarded)
    const long long row = shortlist[(size_t)b * K_ + k];

    const float* wrow = weight + (size_t)row * D_ + (half << 1); // +0 or +2 floats
    const float* eptr = s_e + (half << 1);

    v8f acc = {0.f, 0.f, 0.f, 0.f, 0.f, 0.f, 0.f, 0.f};

#pragma unroll 8
    for (int d0 = 0; d0 < D_; d0 += 4) {
        v2f a  = *(const v2f*)(wrow + d0);   // global_load_b64, 16 rows x 16B / instr
        v2f bb = *(const v2f*)(eptr + d0);   // ds_load_b64 from staged embed
        // D = A(16x4 f32) * B(4x16 f32) + C ; 8 args:
        // (neg_a, A, neg_b, B, c_mod, C, reuse_a, reuse_b)
        acc = __builtin_amdgcn_wmma_f32_16x16x4_f32(
            false, a, false, bb, (short)0, acc, false, false);
    }

    // Column 0 of D: lane 0 holds M=0..7 in acc[0..7], lane 16 holds M=8..15.
    if (lane == 0) {
#pragma unroll
        for (int r = 0; r < 8; ++r) s_col[r] = acc[r];
    } else if (lane == 16) {
#pragma unroll
        for (int r = 0; r < 8; ++r) s_col[8 + r] = acc[r];
    }
    __syncthreads();

    if (half == 0 && (kb + m) < K_) {
        out[(size_t)b * K_ + kb + m] = s_col[m] + bias[row];
    }
}

extern "C" void kernel_launch(void* const* d_in, const int* /*in_sizes*/, int /*n_in*/,
                              void* d_out, int /*out_size*/, void* /*d_ws*/, size_t /*ws_size*/,
                              hipStream_t stream) {
    const float*     embed     = (const float*)d_in[0];
    const long long* shortlist = (const long long*)d_in[1];  // int64 per reference
    const float*     weight    = (const float*)d_in[2];
    const float*     bias      = (const float*)d_in[3];
    float*           out       = (float*)d_out;

    dim3 grid(KT_, B_);   // 32 k-tiles x 256 batch rows = 8192 waves
    sparse_linear_wmma<<<grid, 32, 0, stream>>>(embed, shortlist, weight, bias, out);
}